// Model_8014408974458
// MI455X (gfx1250) — compile-verified
//
#include <hip/hip_runtime.h>
#include <stdint.h>

typedef __attribute__((ext_vector_type(2))) float v2f;
typedef __attribute__((ext_vector_type(8))) float v8f;

#define NNODE 8192
#define NEDGE (32 * 8192)
#define IND   512
#define HIDD  256
#define OUTD  64
#define CBN   8192

// ---------- helpers ----------
__device__ __forceinline__ unsigned fkey(float f) {
  unsigned u = __float_as_uint(f);
  return (u & 0x80000000u) ? ~u : (u | 0x80000000u);
}
__device__ __forceinline__ float funkey(unsigned k) {
  unsigned u = (k & 0x80000000u) ? (k & 0x7FFFFFFFu) : ~k;
  return __uint_as_float(u);
}

// CDNA5 async global->LDS copy, 16B per lane (GV mode), tracked by ASYNCcnt.
__device__ __forceinline__ void async_ld16(float* ldsDst, const float* g) {
  unsigned l = (unsigned)(uintptr_t)ldsDst;  // low 32 bits = LDS byte offset
  unsigned long long ga = (unsigned long long)(uintptr_t)g;
  asm volatile("global_load_async_to_lds_b128 %0, %1, off"
               :: "v"(l), "v"(ga) : "memory");
}
// Partial wait: allow the 8 newest (prefetch) async ops to stay in flight.
__device__ __forceinline__ void async_wait8() {
  asm volatile("s_wait_asynccnt 0x8" ::: "memory");
}
__device__ __forceinline__ void async_wait0() {
  asm volatile("s_wait_asynccnt 0x0" ::: "memory");
}

// ---------- degree / norm prep ----------
__global__ __launch_bounds__(256) void deg_kernel(
    const long long* __restrict__ src, const long long* __restrict__ dst,
    float* __restrict__ degOut, float* __restrict__ degIn) {
  int e = blockIdx.x * 256 + threadIdx.x;
  if (e < NEDGE) {
    atomicAdd(&degOut[(int)src[e]], 1.0f);
    atomicAdd(&degIn[(int)dst[e]], 1.0f);
  }
}

__global__ __launch_bounds__(256) void ninv_kernel(
    const float* __restrict__ degOut, const float* __restrict__ degIn,
    float* __restrict__ nOut, float* __restrict__ nIn) {
  int i = blockIdx.x * 256 + threadIdx.x;
  if (i < NNODE) {
    nOut[i] = 1.0f / sqrtf(fmaxf(degOut[i], 1.0f));
    nIn[i]  = 1.0f / sqrtf(fmaxf(degIn[i], 1.0f));
  }
}

// ---------- NN GEMM: C[M,Nc] = A[M,K] @ B[K,Nc]; 64x64 block, 32x32/wave ----
// Double-buffered async global->LDS staging pipelined against WMMA compute.
template <bool SCALE, bool BIAS>
__global__ __launch_bounds__(128) void gemm_nn(
    const float* __restrict__ A, const float* __restrict__ B,
    float* __restrict__ C, int M, int K, int Nc,
    const float* __restrict__ rowScale, const float* __restrict__ bias) {
  __shared__ alignas(16) float As[2][64][36];  // 144B pitch (16B aligned)
  __shared__ alignas(16) float Bs[2][32][68];  // 272B pitch (16B aligned)
  const int tid = threadIdx.x;
  const int lane = tid & 31, wave = tid >> 5;
  const int wr = wave >> 1, wc = wave & 1;
  const int m = lane & 15, hi = lane >> 4;
  const int rowBase = blockIdx.y * 64, colBase = blockIdx.x * 64;
  v8f acc00 = {}, acc01 = {}, acc10 = {}, acc11 = {};

  auto issue = [&](int k0, int buf) {
#pragma unroll
    for (int i = 0; i < 4; ++i) {
      int f4i = tid + i * 128;                 // 512 float4 slots
      int ra = f4i >> 3, ca = (f4i & 7) << 2;  // A: 64 rows x 32 k
      async_ld16(&As[buf][ra][ca],
                 &A[(size_t)(rowBase + ra) * K + k0 + ca]);
      int rb = f4i >> 4, cb = (f4i & 15) << 2;  // B: 32 k x 64 cols
      async_ld16(&Bs[buf][rb][cb],
                 &B[(size_t)(k0 + rb) * Nc + colBase + cb]);
    }
  };

  const int nchunk = K >> 5;
  issue(0, 0);
  for (int ci = 0; ci < nchunk; ++ci) {
    const int cur = ci & 1;
    const bool more = (ci + 1) < nchunk;
    if (more) issue((ci + 1) << 5, cur ^ 1);  // prefetch next chunk
    if (more) async_wait8(); else async_wait0();
    __syncthreads();
#pragma unroll
    for (int kk = 0; kk < 32; kk += 4) {
      v2f a0, a1, b0, b1;
      const int ar0 = wr * 32 + m, ar1 = ar0 + 16;
      const int kf = kk + 2 * hi;
      a0[0] = As[cur][ar0][kf]; a0[1] = As[cur][ar0][kf + 1];
      a1[0] = As[cur][ar1][kf]; a1[1] = As[cur][ar1][kf + 1];
      const int bc0 = wc * 32 + m, bc1 = bc0 + 16;
      b0[0] = Bs[cur][kf][bc0]; b0[1] = Bs[cur][kf + 1][bc0];
      b1[0] = Bs[cur][kf][bc1]; b1[1] = Bs[cur][kf + 1][bc1];
      acc00 = __builtin_amdgcn_wmma_f32_16x16x4_f32(false, a0, false, b0,
                                                    (short)0, acc00, false, false);
      acc01 = __builtin_amdgcn_wmma_f32_16x16x4_f32(false, a0, false, b1,
                                                    (short)0, acc01, false, false);
      acc10 = __builtin_amdgcn_wmma_f32_16x16x4_f32(false, a1, false, b0,
                                                    (short)0, acc10, false, false);
      acc11 = __builtin_amdgcn_wmma_f32_16x16x4_f32(false, a1, false, b1,
                                                    (short)0, acc11, false, false);
    }
    __syncthreads();
  }
  auto store = [&](v8f& a, int mt, int nt) {
    const int c = colBase + wc * 32 + nt * 16 + m;
    const float bv = BIAS ? bias[c] : 0.0f;
#pragma unroll
    for (int v = 0; v < 8; ++v) {
      const int r = rowBase + wr * 32 + mt * 16 + v + 8 * hi;
      float val = a[v];
      if (SCALE) val *= rowScale[r];
      C[(size_t)r * Nc + c] = val + bv;
    }
  };
  store(acc00, 0, 0); store(acc01, 0, 1);
  store(acc10, 1, 0); store(acc11, 1, 1);
}

// ---------- NT GEMM for dist: D[i,j]=(h[i].cn[j])*rnorm[i]; fused argmax ----
__global__ __launch_bounds__(128) void dist_kernel(
    const float* __restrict__ H, const float* __restrict__ Cn,
    const float* __restrict__ rnorm, float* __restrict__ Dist,
    unsigned long long* __restrict__ rowmax) {
  __shared__ alignas(16) float As[2][64][36];
  __shared__ alignas(16) float Bs[2][64][36];
  const int tid = threadIdx.x;
  const int lane = tid & 31, wave = tid >> 5;
  const int wr = wave >> 1, wc = wave & 1;
  const int m = lane & 15, hi = lane >> 4;
  const int rowBase = blockIdx.y * 64, colBase = blockIdx.x * 64;
  v8f acc00 = {}, acc01 = {}, acc10 = {}, acc11 = {};

  auto issue = [&](int k0, int buf) {
#pragma unroll
    for (int i = 0; i < 4; ++i) {
      int f4i = tid + i * 128;
      int r = f4i >> 3, c4 = (f4i & 7) << 2;
      async_ld16(&As[buf][r][c4], &H[(size_t)(rowBase + r) * IND + k0 + c4]);
      async_ld16(&Bs[buf][r][c4], &Cn[(size_t)(colBase + r) * IND + k0 + c4]);
    }
  };

  const int nchunk = IND >> 5;
  issue(0, 0);
  for (int ci = 0; ci < nchunk; ++ci) {
    const int cur = ci & 1;
    const bool more = (ci + 1) < nchunk;
    if (more) issue((ci + 1) << 5, cur ^ 1);
    if (more) async_wait8(); else async_wait0();
    __syncthreads();
#pragma unroll
    for (int kk = 0; kk < 32; kk += 4) {
      v2f a0, a1, b0, b1;
      const int ar0 = wr * 32 + m, ar1 = ar0 + 16;
      const int bc0 = wc * 32 + m, bc1 = bc0 + 16;
      const int kf = kk + 2 * hi;
      a0[0] = As[cur][ar0][kf]; a0[1] = As[cur][ar0][kf + 1];
      a1[0] = As[cur][ar1][kf]; a1[1] = As[cur][ar1][kf + 1];
      b0[0] = Bs[cur][bc0][kf]; b0[1] = Bs[cur][bc0][kf + 1];  // NT layout
      b1[0] = Bs[cur][bc1][kf]; b1[1] = Bs[cur][bc1][kf + 1];
      acc00 = __builtin_amdgcn_wmma_f32_16x16x4_f32(false, a0, false, b0,
                                                    (short)0, acc00, false, false);
      acc01 = __builtin_amdgcn_wmma_f32_16x16x4_f32(false, a0, false, b1,
                                                    (short)0, acc01, false, false);
      acc10 = __builtin_amdgcn_wmma_f32_16x16x4_f32(false, a1, false, b0,
                                                    (short)0, acc10, false, false);
      acc11 = __builtin_amdgcn_wmma_f32_16x16x4_f32(false, a1, false, b1,
                                                    (short)0, acc11, false, false);
    }
    __syncthreads();
  }
  auto emit = [&](v8f& a, int mt, int nt) {
    const int c = colBase + wc * 32 + nt * 16 + m;
#pragma unroll
    for (int v = 0; v < 8; ++v) {
      const int r = rowBase + wr * 32 + mt * 16 + v + 8 * hi;
      float val = a[v] * rnorm[r];
      Dist[(size_t)r * CBN + c] = val;
      // packed (orderedKey, ~col): max = best value, lowest col on ties
      unsigned long long p = ((unsigned long long)fkey(val) << 32) |
                             (unsigned long long)(0xFFFFFFFFu - (unsigned)c);
#pragma unroll
      for (int s2 = 1; s2 <= 8; s2 <<= 1) {
        unsigned long long o = __shfl_xor(p, s2, 32);
        p = (o > p) ? o : p;
      }
      if (m == 0) atomicMax(&rowmax[r], p);
    }
  };
  emit(acc00, 0, 0); emit(acc01, 0, 1);
  emit(acc10, 1, 0); emit(acc11, 1, 1);
}

// ---------- NT GEMM over qe.qe^T: fused global min/max/sum/sumsq only ------
__global__ __launch_bounds__(128) void adjstat_kernel(
    const float* __restrict__ Qe, double* __restrict__ sumAQ,
    double* __restrict__ sumAQ2, unsigned* __restrict__ minKey,
    unsigned* __restrict__ maxKey) {
  __shared__ alignas(16) float As[2][64][36];
  __shared__ alignas(16) float Bs[2][64][36];
  const int tid = threadIdx.x;
  const int lane = tid & 31, wave = tid >> 5;
  const int wr = wave >> 1, wc = wave & 1;
  const int m = lane & 15, hi = lane >> 4;
  const int rowBase = blockIdx.y * 64, colBase = blockIdx.x * 64;
  v8f acc00 = {}, acc01 = {}, acc10 = {}, acc11 = {};

  auto issue = [&](int k0, int buf) {
#pragma unroll
    for (int i = 0; i < 4; ++i) {
      int f4i = tid + i * 128;
      int r = f4i >> 3, c4 = (f4i & 7) << 2;
      async_ld16(&As[buf][r][c4], &Qe[(size_t)(rowBase + r) * IND + k0 + c4]);
      async_ld16(&Bs[buf][r][c4], &Qe[(size_t)(colBase + r) * IND + k0 + c4]);
    }
  };

  const int nchunk = IND >> 5;
  issue(0, 0);
  for (int ci = 0; ci < nchunk; ++ci) {
    const int cur = ci & 1;
    const bool more = (ci + 1) < nchunk;
    if (more) issue((ci + 1) << 5, cur ^ 1);
    if (more) async_wait8(); else async_wait0();
    __syncthreads();
#pragma unroll
    for (int kk = 0; kk < 32; kk += 4) {
      v2f a0, a1, b0, b1;
      const int ar0 = wr * 32 + m, ar1 = ar0 + 16;
      const int bc0 = wc * 32 + m, bc1 = bc0 + 16;
      const int kf = kk + 2 * hi;
      a0[0] = As[cur][ar0][kf]; a0[1] = As[cur][ar0][kf + 1];
      a1[0] = As[cur][ar1][kf]; a1[1] = As[cur][ar1][kf + 1];
      b0[0] = Bs[cur][bc0][kf]; b0[1] = Bs[cur][bc0][kf + 1];
      b1[0] = Bs[cur][bc1][kf]; b1[1] = Bs[cur][bc1][kf + 1];
      acc00 = __builtin_amdgcn_wmma_f32_16x16x4_f32(false, a0, false, b0,
                                                    (short)0, acc00, false, false);
      acc01 = __builtin_amdgcn_wmma_f32_16x16x4_f32(false, a0, false, b1,
                                                    (short)0, acc01, false, false);
      acc10 = __builtin_amdgcn_wmma_f32_16x16x4_f32(false, a1, false, b0,
                                                    (short)0, acc10, false, false);
      acc11 = __builtin_amdgcn_wmma_f32_16x16x4_f32(false, a1, false, b1,
                                                    (short)0, acc11, false, false);
    }
    __syncthreads();
  }
  float lmin = 3.4e38f, lmax = -3.4e38f;
  double lsum = 0.0, lsq = 0.0;
  auto reduceAcc = [&](v8f& a) {
#pragma unroll
    for (int v = 0; v < 8; ++v) {
      float val = a[v];
      lmin = fminf(lmin, val);
      lmax = fmaxf(lmax, val);
      lsum += (double)val;
      lsq += (double)val * (double)val;
    }
  };
  reduceAcc(acc00); reduceAcc(acc01); reduceAcc(acc10); reduceAcc(acc11);
#pragma unroll
  for (int s2 = 1; s2 < 32; s2 <<= 1) {
    lmin = fminf(lmin, __shfl_xor(lmin, s2, 32));
    lmax = fmaxf(lmax, __shfl_xor(lmax, s2, 32));
    lsum += __shfl_xor(lsum, s2, 32);
    lsq += __shfl_xor(lsq, s2, 32);
  }
  if (lane == 0) {
    atomicMin(minKey, fkey(lmin));
    atomicMax(maxKey, fkey(lmax));
    atomicAdd(sumAQ, lsum);
    atomicAdd(sumAQ2, lsq);
  }
}

// ---------- scatter-add aggregation: agg[dst] += x[src] ----------
__global__ __launch_bounds__(256) void scatter_add(
    const float* __restrict__ X, float* __restrict__ Agg,
    const long long* __restrict__ src, const long long* __restrict__ dst, int dim) {
  const int per = dim >> 2;
  const long long total = (long long)NEDGE * per;
  for (long long t = (long long)blockIdx.x * 256 + threadIdx.x; t < total;
       t += (long long)gridDim.x * 256) {
    long long e = t / per;
    int c4 = (int)(t - e * per) << 2;
    int s = (int)src[e], d = (int)dst[e];
    float4 v = *(const float4*)&X[(size_t)s * dim + c4];
    float* p = &Agg[(size_t)d * dim + c4];
    atomicAdd(p + 0, v.x);
    atomicAdd(p + 1, v.y);
    atomicAdd(p + 2, v.z);
    atomicAdd(p + 3, v.w);
  }
}

// ---------- h = relu(agg * nIn[row] + b[col]) ----------
__global__ __launch_bounds__(256) void postagg_kernel(
    const float* __restrict__ agg, const float* __restrict__ nIn,
    const float* __restrict__ bias, float* __restrict__ out, int dim) {
  const long long total = (long long)NNODE * dim;
  for (long long t = (long long)blockIdx.x * 256 + threadIdx.x; t < total;
       t += (long long)gridDim.x * 256) {
    int row = (int)(t / dim);
    int c = (int)(t - (long long)row * dim);
    out[t] = fmaxf(agg[t] * nIn[row] + bias[c], 0.0f);
  }
}

// ---------- per-row inverse norms ----------
__global__ __launch_bounds__(32) void rownorm_kernel(
    const float* __restrict__ H, float* __restrict__ rnorm) {
  int i = blockIdx.x, lane = threadIdx.x;
  const float* r = &H[(size_t)i * IND];
  float s = 0.0f;
  for (int k = lane; k < IND; k += 32) s += r[k] * r[k];
#pragma unroll
  for (int d = 16; d; d >>= 1) s += __shfl_xor(s, d, 32);
  if (lane == 0) rnorm[i] = 1.0f / (sqrtf(s) + 1e-12f);
}

__global__ __launch_bounds__(32) void cnorm_kernel(
    const float* __restrict__ CBk, float* __restrict__ Cn) {
  int i = blockIdx.x, lane = threadIdx.x;
  const float* r = &CBk[(size_t)i * IND];
  float s = 0.0f;
  for (int k = lane; k < IND; k += 32) s += r[k] * r[k];
#pragma unroll
  for (int d = 16; d; d >>= 1) s += __shfl_xor(s, d, 32);
  float inv = 1.0f / (sqrtf(s) + 1e-12f);
  float* o = &Cn[(size_t)i * IND];
  for (int k = lane; k < IND; k += 32) o[k] = r[k] * inv;
}

// ---------- decode argmax, gather q=cn[ind], accumulate commit loss --------
__global__ __launch_bounds__(128) void qgather_commit(
    const float* __restrict__ H, const float* __restrict__ Cn,
    const unsigned long long* __restrict__ rowmax, float* __restrict__ Q,
    double* __restrict__ commitSum) {
  __shared__ double red[128];
  int i = blockIdx.x;
  unsigned idx = 0xFFFFFFFFu - (unsigned)(rowmax[i] & 0xFFFFFFFFull);
  const float* c = &Cn[(size_t)idx * IND];
  const float* h = &H[(size_t)i * IND];
  float* q = &Q[(size_t)i * IND];
  double loc = 0.0;
  for (int d = threadIdx.x; d < IND; d += 128) {
    float qa = c[d];
    q[d] = qa;
    float df = qa - h[d];
    loc += (double)df * df;
  }
  red[threadIdx.x] = loc;
  __syncthreads();
  for (int s = 64; s; s >>= 1) {
    if (threadIdx.x < (unsigned)s) red[threadIdx.x] += red[threadIdx.x + s];
    __syncthreads();
  }
  if (threadIdx.x == 0) atomicAdd(commitSum, red[0]);
}

// ---------- feature reconstruction loss partials ----------
__global__ __launch_bounds__(256) void fr_reduce(
    const float* __restrict__ H, const float* __restrict__ Qn,
    double* __restrict__ frSum) {
  __shared__ double red[256];
  const long long total = (long long)NNODE * IND;
  double loc = 0.0;
  for (long long t = (long long)blockIdx.x * 256 + threadIdx.x; t < total;
       t += (long long)gridDim.x * 256) {
    float df = H[t] - Qn[t];
    loc += (double)df * df;
  }
  red[threadIdx.x] = loc;
  __syncthreads();
  for (int s = 128; s; s >>= 1) {
    if (threadIdx.x < (unsigned)s) red[threadIdx.x] += red[threadIdx.x + s];
    __syncthreads();
  }
  if (threadIdx.x == 0) atomicAdd(frSum, red[0]);
}

// ---------- dedup'd edge cells: sum adj_q at edges + count ----------
__global__ __launch_bounds__(256) void edge_stats_kernel(
    const long long* __restrict__ src, const long long* __restrict__ dst,
    const float* __restrict__ Qe, unsigned* __restrict__ mask,
    double* __restrict__ edgeSum, unsigned* __restrict__ edgeCnt) {
  int wave = threadIdx.x >> 5, lane = threadIdx.x & 31;
  long long e = (long long)blockIdx.x * 8 + wave;
  if (e >= NEDGE) return;
  int s = (int)src[e], d = (int)dst[e];
  int owner = 0;
  if (lane == 0) {
    unsigned bitpos = (unsigned)s * (unsigned)NNODE + (unsigned)d;
    unsigned old = atomicOr(&mask[bitpos >> 5], 1u << (bitpos & 31));
    owner = ((old >> (bitpos & 31)) & 1u) == 0u;
  }
  owner = __shfl(owner, 0, 32);
  if (!owner) return;
  const float* a = &Qe[(size_t)s * IND];
  const float* b = &Qe[(size_t)d * IND];
  float acc = 0.0f;
  for (int k = lane; k < IND; k += 32) acc += a[k] * b[k];
#pragma unroll
  for (int s2 = 16; s2; s2 >>= 1) acc += __shfl_xor(acc, s2, 32);
  if (lane == 0) {
    atomicAdd(edgeSum, (double)acc);
    atomicAdd(edgeCnt, 1u);
  }
}

// ---------- init / finalize ----------
__global__ void init_stats(unsigned* uS) { uS[0] = 0xFFFFFFFFu; }

__global__ void finalize_kernel(const double* __restrict__ dS,
                                const unsigned* __restrict__ uS,
                                float* __restrict__ lossOut) {
  // dS: [commit, fr, sumAQ, sumAQ2, edgeSum]; uS: [minKey, maxKey, edgeCnt]
  double denomNF = (double)NNODE * (double)IND;
  double commit = 0.25 * dS[0] / denomNF;
  double fr = 0.001 * dS[1] / denomNF;
  double mn = (double)funkey(uS[0]);
  double mx = (double)funkey(uS[1]);
  double M = (double)NNODE * (double)NNODE;
  double inv = 1.0 / (mx - mn);
  double sum_c2 = (dS[3] - 2.0 * mn * dS[2] + M * mn * mn) * inv * inv;
  double cnt = (double)uS[2];
  double sum_edge_c = (dS[4] - cnt * mn) * inv;
  double S = sum_c2 - 2.0 * sum_edge_c + cnt;  // sum (adj - adj_q_norm)^2
  double edge = 0.03 * sqrt(S / M);
  lossOut[0] = (float)(fr + edge + commit);
}

// ======================================================================
extern "C" void kernel_launch(void* const* d_in, const int* in_sizes, int n_in,
                              void* d_out, int out_size, void* d_ws,
                              size_t ws_size, hipStream_t stream) {
  (void)in_sizes; (void)n_in; (void)out_size; (void)ws_size;
  const float* feats = (const float*)d_in[0];
  const long long* eidx = (const long long*)d_in[1];  // int64 [2,E]
  const float* W1 = (const float*)d_in[2];
  const float* b1 = (const float*)d_in[3];
  const float* W2 = (const float*)d_in[4];
  const float* b2 = (const float*)d_in[5];
  const float* dW1 = (const float*)d_in[6];
  const float* db1 = (const float*)d_in[7];
  const float* dW2 = (const float*)d_in[8];
  const float* db2 = (const float*)d_in[9];
  const float* linW = (const float*)d_in[10];
  const float* linb = (const float*)d_in[11];
  const float* codebook = (const float*)d_in[12];
  const long long* src = eidx;
  const long long* dst = eidx + NEDGE;

  char* ws = (char*)d_ws;
  const size_t MB = 1024 * 1024;
  float* h    = (float*)(ws + 0 * MB);    // N x 512
  float* cn   = (float*)(ws + 16 * MB);   // CB x 512 normalized codebook
  float* bufA = (float*)(ws + 32 * MB);   // xw1, then qmat
  float* bufB = (float*)(ws + 48 * MB);   // agg1, then qe
  float* qn   = (float*)(ws + 64 * MB);   // N x 512
  float* xw2  = (float*)(ws + 80 * MB);   // N x 256
  float* agg2 = (float*)(ws + 88 * MB);   // N x 256
  float* h2   = (float*)(ws + 96 * MB);   // N x 256
  unsigned* mask = (unsigned*)(ws + 104 * MB);  // N*N bits = 8MB
  float* degOut = (float*)(ws + 112 * MB);
  float* degIn = degOut + NNODE;
  float* nOut = degIn + NNODE;
  float* nIn = nOut + NNODE;
  float* rnorm = nIn + NNODE;
  unsigned long long* rowmax = (unsigned long long*)((char*)(rnorm + NNODE));
  double* dS = (double*)((char*)rowmax + (size_t)NNODE * 8);
  unsigned* uS = (unsigned*)(dS + 5);

  float* outMat = (float*)d_out;                 // [N, 64]
  float* lossOut = outMat + (size_t)NNODE * OUTD;
  float* distOut = lossOut + 1;                  // [N, CB]

  // zero accumulators / masks (capturable async memsets)
  hipMemsetAsync(bufB, 0, (size_t)NNODE * IND * 4, stream);   // agg1
  hipMemsetAsync(agg2, 0, (size_t)NNODE * HIDD * 4, stream);
  hipMemsetAsync(mask, 0, (size_t)NNODE * NNODE / 8, stream);
  hipMemsetAsync(degOut, 0, (size_t)2 * NNODE * 4, stream);
  hipMemsetAsync(rowmax, 0, (size_t)NNODE * 8, stream);
  hipMemsetAsync(dS, 0, 5 * 8 + 3 * 4, stream);
  init_stats<<<1, 1, 0, stream>>>(uS);

  // degrees + normalizers
  deg_kernel<<<(NEDGE + 255) / 256, 256, 0, stream>>>(src, dst, degOut, degIn);
  ninv_kernel<<<(NNODE + 255) / 256, 256, 0, stream>>>(degOut, degIn, nOut, nIn);

  // conv1: xw1 = (feats @ W1) * nOut
  gemm_nn<true, false><<<dim3(IND / 64, NNODE / 64), 128, 0, stream>>>(
      feats, W1, bufA, NNODE, IND, IND, nOut, nullptr);
  scatter_add<<<4096, 256, 0, stream>>>(bufA, bufB, src, dst, IND);
  postagg_kernel<<<4096, 256, 0, stream>>>(bufB, nIn, b1, h, IND);

  // normalization for cosine sim
  rownorm_kernel<<<NNODE, 32, 0, stream>>>(h, rnorm);
  cnorm_kernel<<<CBN, 32, 0, stream>>>(codebook, cn);

  // dist = hn @ cn^T (fused row-scale + argmax), writes d_out dist block
  dist_kernel<<<dim3(CBN / 64, NNODE / 64), 128, 0, stream>>>(
      h, cn, rnorm, distOut, rowmax);

  // quantized = cn[argmax] (straight-through forward value) + commit loss
  qgather_commit<<<NNODE, 128, 0, stream>>>(h, cn, rowmax, bufA, &dS[0]);

  // quantized_edge / quantized_node
  gemm_nn<false, true><<<dim3(IND / 64, NNODE / 64), 128, 0, stream>>>(
      bufA, dW1, bufB, NNODE, IND, IND, nullptr, db1);   // qe
  gemm_nn<false, true><<<dim3(IND / 64, NNODE / 64), 128, 0, stream>>>(
      bufA, dW2, qn, NNODE, IND, IND, nullptr, db2);     // qn
  fr_reduce<<<4096, 256, 0, stream>>>(h, qn, &dS[1]);

  // edge reconstruction loss statistics (never materializes adj / adj_q)
  adjstat_kernel<<<dim3(NNODE / 64, NNODE / 64), 128, 0, stream>>>(
      bufB, &dS[2], &dS[3], &uS[0], &uS[1]);
  edge_stats_kernel<<<NEDGE / 8, 256, 0, stream>>>(
      src, dst, bufB, mask, &dS[4], &uS[2]);

  // conv2: xw2 = (qe @ W2) * nOut ; aggregate ; relu
  gemm_nn<true, false><<<dim3(HIDD / 64, NNODE / 64), 128, 0, stream>>>(
      bufB, W2, xw2, NNODE, IND, HIDD, nOut, nullptr);
  scatter_add<<<4096, 256, 0, stream>>>(xw2, agg2, src, dst, HIDD);
  postagg_kernel<<<4096, 256, 0, stream>>>(agg2, nIn, b2, h2, HIDD);

  // out = h2 @ linW + linb
  gemm_nn<false, true><<<dim3(OUTD / 64, NNODE / 64), 128, 0, stream>>>(
      h2, linW, outMat, NNODE, HIDD, OUTD, nullptr, linb);

  finalize_kernel<<<1, 1, 0, stream>>>(dS, uS, lossOut);
}